// PointCloudLoss_22789096472822
// MI455X (gfx1250) — compile-verified
//
#include <hip/hip_runtime.h>
#include <hip/hip_bf16.h>
#include <math.h>
#include <stdint.h>

#define BB 16
#define NN 1024
#define DD 3
#define EPSF 1e-8f
#define FMAXV 3.402823466e+38f
#define EXP_NEG10 4.539992976e-5f
#define EXP_POS10 22026.465795f

typedef __attribute__((ext_vector_type(2))) float v2f;
typedef __attribute__((ext_vector_type(8))) float v8f;
typedef __attribute__((ext_vector_type(4))) int v4i;

typedef __attribute__((address_space(1))) v4i* gv4i_p;  // global
typedef __attribute__((address_space(3))) v4i* lv4i_p;  // LDS

#ifndef __has_builtin
#define __has_builtin(x) 0
#endif
#if __has_builtin(__builtin_amdgcn_global_load_async_to_lds_b128) && \
    __has_builtin(__builtin_amdgcn_s_wait_asynccnt)
#define HAVE_ASYNC_LDS 1
#else
#define HAVE_ASYNC_LDS 0
#endif

// ---------------- kernel 0: zero scalar accumulators ----------------
__global__ void pcl_init_accum(float* accum) {
    if (threadIdx.x < 4) accum[threadIdx.x] = 0.0f;
}

// ---------------- kernel 1: normalize points, init mins, MSE + EMD partials ----
__global__ __launch_bounds__(256)
void pcl_prep(const float* __restrict__ noise_pred,
              const float* __restrict__ noise_target,
              const float* __restrict__ pred,
              const float* __restrict__ target,
              const int* __restrict__ assign,
              float4* __restrict__ xn, float4* __restrict__ yn,
              float* __restrict__ rowmin, float* __restrict__ colmin,
              float* __restrict__ accum) {
    int p = blockIdx.x * blockDim.x + threadIdx.x; // 0 .. B*N-1
    float mse = 0.0f, emd = 0.0f;
    if (p < BB * NN) {
        int b = p / NN;
        // diffusion MSE partial
        #pragma unroll
        for (int d = 0; d < DD; ++d) {
            float df = noise_pred[p * DD + d] - noise_target[p * DD + d];
            mse += df * df;
        }
        // chamfer: clip [-1,1], normalize, pad to float4 (w=0)
        float px = fminf(fmaxf(pred[p * DD + 0], -1.0f), 1.0f);
        float py = fminf(fmaxf(pred[p * DD + 1], -1.0f), 1.0f);
        float pz = fminf(fmaxf(pred[p * DD + 2], -1.0f), 1.0f);
        float pn = fmaxf(sqrtf(px * px + py * py + pz * pz), EPSF);
        xn[p] = make_float4(px / pn, py / pn, pz / pn, 0.0f);

        float tx = fminf(fmaxf(target[p * DD + 0], -1.0f), 1.0f);
        float ty = fminf(fmaxf(target[p * DD + 1], -1.0f), 1.0f);
        float tz = fminf(fmaxf(target[p * DD + 2], -1.0f), 1.0f);
        float tn = fmaxf(sqrtf(tx * tx + ty * ty + tz * tz), EPSF);
        yn[p] = make_float4(tx / tn, ty / tn, tz / tn, 0.0f);

        rowmin[p] = FMAXV;
        colmin[p] = FMAXV;

        // EMD: clip [-2,2], matched by assign (stop_gradient irrelevant for fwd value)
        int a = assign[p];
        int q = b * NN + a;
        float ex = fminf(fmaxf(pred[p * DD + 0], -2.0f), 2.0f);
        float ey = fminf(fmaxf(pred[p * DD + 1], -2.0f), 2.0f);
        float ez = fminf(fmaxf(pred[p * DD + 2], -2.0f), 2.0f);
        float fx = fminf(fmaxf(target[q * DD + 0], -2.0f), 2.0f);
        float fy = fminf(fmaxf(target[q * DD + 1], -2.0f), 2.0f);
        float fz = fminf(fmaxf(target[q * DD + 2], -2.0f), 2.0f);
        float dx = ex - fx, dy = ey - fy, dz = ez - fz;
        emd = fmaxf(sqrtf(dx * dx + dy * dy + dz * dz), EPSF);
    }
    __shared__ float smse[256];
    __shared__ float semd[256];
    int tid = threadIdx.x;
    smse[tid] = mse;
    semd[tid] = emd;
    __syncthreads();
    for (int s = 128; s > 0; s >>= 1) {
        if (tid < s) { smse[tid] += smse[tid + s]; semd[tid] += semd[tid + s]; }
        __syncthreads();
    }
    if (tid == 0) {
        atomicAdd(&accum[0], smse[0]);
        atomicAdd(&accum[1], semd[0]);
    }
}

// ---------------- kernel 2: WMMA pairwise dots -> squared-dist mins ----------
// grid = B * (N/16) / 4 = 256 blocks, 128 threads (4 waves). Each wave owns one
// 16-row tile of batch b; block stages yn[b] (16KB) in LDS (async if available);
// 64 column tiles, one v_wmma_f32_16x16x4_f32 each (K=4, D padded), unrolled x4.
// Col-min partials go to LDS (no divergence in the hot loop); one atomic per
// column per block in the epilogue.
__global__ __launch_bounds__(128)
void pcl_chamfer_wmma(const float4* __restrict__ xn,
                      const float4* __restrict__ yn,
                      float* __restrict__ rowmin,
                      unsigned int* __restrict__ colmin) {
    __shared__ float4 ylds[NN];              // 16 KB: this batch's yn
    __shared__ float colpart[8 * NN];        // 32 KB: [wave*2+half][col] partial mins

    int b = blockIdx.x >> 4;                 // 16 blocks per batch
    int tid = threadIdx.x;

#if HAVE_ASYNC_LDS
    {
        // gfx1250 async global->LDS copy (ASYNCcnt-tracked), 16B per lane per call
        uintptr_t gsrc = (uintptr_t)(const void*)(yn + (size_t)b * NN);
        unsigned lbase = (unsigned)(uintptr_t)(void*)&ylds[0];  // low 32 bits = LDS offset
        for (int i = tid; i < NN; i += 128) {
            __builtin_amdgcn_global_load_async_to_lds_b128(
                (gv4i_p)(gsrc + (size_t)i * 16),
                (lv4i_p)(lbase + (unsigned)i * 16u),
                0, 0);
        }
        __builtin_amdgcn_s_wait_asynccnt(0);
    }
#else
    for (int i = tid; i < NN; i += 128) {
        ylds[i] = yn[b * NN + i];
    }
#endif
    __syncthreads();

    int lane = tid & 31;
    int waveId = tid >> 5;
    int tile_i = ((blockIdx.x & 15) << 2) + waveId; // 0..63
    int hi = lane >> 4;   // half-wave: K-pair select
    int lr = lane & 15;   // M (for A) / N (for B/D)
    int cpbase = ((waveId << 1) + hi) << 10;        // colpart row base

    // A fragment: 16x4 f32, lane lr holds row M=lr; hi selects K={0,1} vs {2,3}
    const float2* ap = reinterpret_cast<const float2*>(&xn[b * NN + tile_i * 16 + lr]);
    float2 at = ap[hi];
    v2f afrag; afrag[0] = at.x; afrag[1] = at.y;

    float rmin[8];
    #pragma unroll
    for (int r = 0; r < 8; ++r) rmin[r] = FMAXV;

    #pragma unroll 4
    for (int j = 0; j < NN / 16; ++j) {
        // B fragment: 4x16 f32, lane lr holds col N=lr; hi selects K pair
        const float2* bp = reinterpret_cast<const float2*>(&ylds[j * 16 + lr]);
        float2 bt = bp[hi];
        v2f bfrag; bfrag[0] = bt.x; bfrag[1] = bt.y;

        v8f c = {0.f, 0.f, 0.f, 0.f, 0.f, 0.f, 0.f, 0.f};
        v8f dacc = __builtin_amdgcn_wmma_f32_16x16x4_f32(
            /*neg_a=*/false, afrag, /*neg_b=*/false, bfrag,
            /*c_mod=*/(short)0, c, /*reuse_a=*/false, /*reuse_b=*/false);

        // unit vectors: dist^2 = 2 - 2*dot  (clamp at 0)
        float cmin = FMAXV;
        #pragma unroll
        for (int r = 0; r < 8; ++r) {
            float s = fmaxf(2.0f - 2.0f * dacc[r], 0.0f);
            rmin[r] = fminf(rmin[r], s);
            cmin = fminf(cmin, s);
        }
        // store this lane's 8-row partial min for its column; no divergence
        colpart[cpbase + (j << 4) + lr] = cmin;
    }

    // row mins: reduce across the 16 lanes of each half (all N columns seen)
    #pragma unroll
    for (int r = 0; r < 8; ++r) {
        float v = rmin[r];
        v = fminf(v, __shfl_xor(v, 1, 32));
        v = fminf(v, __shfl_xor(v, 2, 32));
        v = fminf(v, __shfl_xor(v, 4, 32));
        v = fminf(v, __shfl_xor(v, 8, 32));
        if (lr == 0) {
            // D layout: VGPR r, half hi -> M = r + 8*hi
            rowmin[b * NN + tile_i * 16 + hi * 8 + r] = v;
        }
    }

    // col mins: combine the 8 partials (4 waves x 2 halves), 1 atomic per column
    __syncthreads();
    for (int col = tid; col < NN; col += 128) {
        float m = colpart[col];
        #pragma unroll
        for (int k = 1; k < 8; ++k) {
            m = fminf(m, colpart[(k << 10) + col]);
        }
        // positive floats: uint bit order == float order
        atomicMin(&colmin[b * NN + col], __float_as_uint(m));
    }
}

// ---------------- kernel 3: finalize scalar --------------------------------
__global__ __launch_bounds__(256)
void pcl_finalize(const float* __restrict__ rowmin,
                  const unsigned int* __restrict__ colmin,
                  const float* __restrict__ accum,
                  float* __restrict__ out) {
    __shared__ float sdl[256];
    __shared__ float sdr[256];
    int tid = threadIdx.x;
    float dl = 0.0f, dr = 0.0f;
    for (int i = tid; i < BB * NN; i += 256) {
        // exp(clip(log(d+eps),-10,10)) == clamp(d+eps, e^-10, e^10); min commutes
        float s = rowmin[i];
        float d = fmaxf(sqrtf(fmaxf(s, 0.0f)), EPSF);
        dl += fminf(fmaxf(d + EPSF, EXP_NEG10), EXP_POS10);

        s = __uint_as_float(colmin[i]);
        d = fmaxf(sqrtf(fmaxf(s, 0.0f)), EPSF);
        dr += fminf(fmaxf(d + EPSF, EXP_NEG10), EXP_POS10);
    }
    sdl[tid] = dl;
    sdr[tid] = dr;
    __syncthreads();
    for (int s = 128; s > 0; s >>= 1) {
        if (tid < s) { sdl[tid] += sdl[tid + s]; sdr[tid] += sdr[tid + s]; }
        __syncthreads();
    }
    if (tid == 0) {
        float dlm = sdl[0] / (float)(BB * NN);
        float drm = sdr[0] / (float)(BB * NN);
        float cd = 0.5f * (dlm + drm);
        float diffusion = accum[0] / (float)(BB * NN * DD);
        float emd = accum[1] / (float)(BB * NN);
        out[0] = 1.0f * diffusion + 0.1f * cd + 0.05f * emd;
    }
}

// ---------------- host launcher --------------------------------------------
extern "C" void kernel_launch(void* const* d_in, const int* in_sizes, int n_in,
                              void* d_out, int out_size, void* d_ws, size_t ws_size,
                              hipStream_t stream) {
    const float* noise_pred   = (const float*)d_in[0];
    const float* noise_target = (const float*)d_in[1];
    const float* pred_points  = (const float*)d_in[2];
    const float* target_points= (const float*)d_in[3];
    const int*   assign       = (const int*)d_in[4];
    float* out = (float*)d_out;

    // workspace layout
    float4* xn = (float4*)d_ws;                                   // B*N float4
    float4* yn = xn + BB * NN;                                    // B*N float4
    float* rowmin = (float*)(yn + BB * NN);                       // B*N f32
    unsigned int* colmin = (unsigned int*)(rowmin + BB * NN);     // B*N u32-encoded f32
    float* accum = (float*)(colmin + BB * NN);                    // 4 f32

    pcl_init_accum<<<1, 32, 0, stream>>>(accum);
    pcl_prep<<<(BB * NN) / 256, 256, 0, stream>>>(
        noise_pred, noise_target, pred_points, target_points, assign,
        xn, yn, rowmin, (float*)colmin, accum);
    pcl_chamfer_wmma<<<BB * (NN / 16) / 4, 128, 0, stream>>>(xn, yn, rowmin, colmin);
    pcl_finalize<<<1, 256, 0, stream>>>(rowmin, colmin, accum, out);
}